// GraphPolicyNet_53102975647980
// MI455X (gfx1250) — compile-verified
//
#include <hip/hip_runtime.h>
#include <hip/hip_bf16.h>

typedef float v2f __attribute__((ext_vector_type(2)));
typedef float v8f __attribute__((ext_vector_type(8)));

__device__ __forceinline__ float lrelu(float v) { return v > 0.0f ? v : 0.01f * v; }

// ---------------------------------------------------------------------------
// WMMA tile convention (V_WMMA_F32_16X16X4_F32, wave32):
//   A (16x4 f32):  lane&15 = row M, lane>=16 selects K+2, v2f = {K, K+1}
//   B (4x16 f32):  lane&15 = col N, lane>=16 selects K+2, v2f = {K, K+1}
//   C/D (16x16):   VGPR v: lanes 0-15 -> M=v, lanes 16-31 -> M=v+8; N = lane&15
//
// Weights (K x 32) are staged into LDS in the per-lane B layout:
//   lds[2*(kk*64 + t*32 + lane) + j] = W[(kk*4 + (lane>>4)*2 + j)*32 + (lane&15) + 16*t]
// Reads are conflict-free b64 DS loads (lane l -> banks 2l, 2l+1).
// ---------------------------------------------------------------------------

template <int NK>
__device__ __forceinline__ void stage_B_lds(const float* __restrict__ W,
                                            float* __restrict__ lds) {
  for (int i = threadIdx.x; i < NK * 64; i += blockDim.x) {
    const int lane_i = i & 31;
    const int t      = (i >> 5) & 1;
    const int kk     = i >> 6;
    const int k      = kk * 4 + ((lane_i >> 4) * 2);
    const int n      = (lane_i & 15) + t * 16;
    lds[2 * i]     = W[(size_t)k * 32 + n];
    lds[2 * i + 1] = W[(size_t)(k + 1) * 32 + n];
  }
  __syncthreads();
}

// x = lrelu(node_feats(N,128) @ W(128,32) + b)
__global__ void __launch_bounds__(256)
gpn_embed_kernel(const float* __restrict__ feats,
                 const float* __restrict__ W,
                 const float* __restrict__ bias,
                 float* __restrict__ x, int N) {
  __shared__ float ldsW[32 * 128];   // 32 k-steps x 2 tiles x 32 lanes x 2 = 16 KB
  stage_B_lds<32>(W, ldsW);

  const int lane = threadIdx.x & 31;
  const int wave = threadIdx.x >> 5;
  const int wpb  = blockDim.x >> 5;
  const int nwaves = gridDim.x * wpb;
  const int ntiles = (N + 15) >> 4;
  const int m    = lane & 15;
  const int koff = (lane >> 4) * 2;
  const int mrow = (lane >> 4) * 8;
  const float bn0 = bias[m], bn1 = bias[16 + m];

  for (int tile = blockIdx.x * wpb + wave; tile < ntiles; tile += nwaves) {
    int rowA = tile * 16 + m;
    if (rowA >= N) rowA = 0;
    const float* arow = feats + (size_t)rowA * 128;
    v8f acc0 = {0.f,0.f,0.f,0.f,0.f,0.f,0.f,0.f};
    v8f acc1 = {0.f,0.f,0.f,0.f,0.f,0.f,0.f,0.f};
#pragma unroll
    for (int kk = 0; kk < 32; ++kk) {
      const int k = kk * 4 + koff;
      v2f a; a.x = arow[k]; a.y = arow[k + 1];
      const v2f b0 = *(const v2f*)(ldsW + 2 * (kk * 64 + lane));
      const v2f b1 = *(const v2f*)(ldsW + 2 * (kk * 64 + 32 + lane));
      acc0 = __builtin_amdgcn_wmma_f32_16x16x4_f32(false, a, false, b0, (short)0, acc0, false, false);
      acc1 = __builtin_amdgcn_wmma_f32_16x16x4_f32(false, a, false, b1, (short)0, acc1, false, false);
    }
#pragma unroll
    for (int v = 0; v < 8; ++v) {
      int node = tile * 16 + v + mrow;
      if (node < N) {
        x[(size_t)node * 32 + m]      = lrelu(acc0[v] + bn0);
        x[(size_t)node * 32 + 16 + m] = lrelu(acc1[v] + bn1);
      }
    }
  }
}

// Fused edge phase: agg[dst] += lrelu(x[src] @ W_msg(32,32) + b_msg)
__global__ void __launch_bounds__(256)
gpn_msg_kernel(const float* __restrict__ x,
               const int* __restrict__ src,
               const int* __restrict__ dst,
               const float* __restrict__ W,
               const float* __restrict__ bias,
               float* __restrict__ agg, int E, int N) {
  __shared__ float ldsW[8 * 128];    // 4 KB
  stage_B_lds<8>(W, ldsW);

  const int lane = threadIdx.x & 31;
  const int wave = threadIdx.x >> 5;
  const int wpb  = blockDim.x >> 5;
  const int nwaves = gridDim.x * wpb;
  const int ntiles = (E + 15) >> 4;
  const int m    = lane & 15;
  const int koff = (lane >> 4) * 2;
  const int mrow = (lane >> 4) * 8;
  const float bn0 = bias[m], bn1 = bias[16 + m];

  for (int tile = blockIdx.x * wpb + wave; tile < ntiles; tile += nwaves) {
    const int e = tile * 16 + m;
    int rowA = (e < E) ? src[e] : 0;
    if ((unsigned)rowA >= (unsigned)N) rowA = 0;   // defensive clamp
    const float* arow = x + (size_t)rowA * 32;

    // Prefetch full A row (16 floats/lane) so loads issue as one clause.
    v2f a[8];
#pragma unroll
    for (int kk = 0; kk < 8; ++kk) {
      const int k = kk * 4 + koff;
      a[kk].x = arow[k]; a[kk].y = arow[k + 1];
    }
    v8f acc0 = {0.f,0.f,0.f,0.f,0.f,0.f,0.f,0.f};
    v8f acc1 = {0.f,0.f,0.f,0.f,0.f,0.f,0.f,0.f};
#pragma unroll
    for (int kk = 0; kk < 8; ++kk) {
      const v2f b0 = *(const v2f*)(ldsW + 2 * (kk * 64 + lane));
      const v2f b1 = *(const v2f*)(ldsW + 2 * (kk * 64 + 32 + lane));
      acc0 = __builtin_amdgcn_wmma_f32_16x16x4_f32(false, a[kk], false, b0, (short)0, acc0, false, false);
      acc1 = __builtin_amdgcn_wmma_f32_16x16x4_f32(false, a[kk], false, b1, (short)0, acc1, false, false);
    }
#pragma unroll
    for (int v = 0; v < 8; ++v) {
      const int eo = tile * 16 + v + mrow;
      if (eo < E) {
        const int d = dst[eo];
        if ((unsigned)d < (unsigned)N) {
          unsafeAtomicAdd(&agg[(size_t)d * 32 + m],      lrelu(acc0[v] + bn0));
          unsafeAtomicAdd(&agg[(size_t)d * 32 + 16 + m], lrelu(acc1[v] + bn1));
        }
      }
    }
  }
}

// x' = lrelu(concat([x, agg], axis=1)(N,64) @ W_upd(64,32) + b_upd)
__global__ void __launch_bounds__(256)
gpn_upd_kernel(const float* __restrict__ x,
               const float* __restrict__ agg,
               const float* __restrict__ W,
               const float* __restrict__ bias,
               float* __restrict__ xout, int N) {
  __shared__ float ldsW[16 * 128];   // 8 KB
  stage_B_lds<16>(W, ldsW);

  const int lane = threadIdx.x & 31;
  const int wave = threadIdx.x >> 5;
  const int wpb  = blockDim.x >> 5;
  const int nwaves = gridDim.x * wpb;
  const int ntiles = (N + 15) >> 4;
  const int m    = lane & 15;
  const int koff = (lane >> 4) * 2;
  const int mrow = (lane >> 4) * 8;
  const float bn0 = bias[m], bn1 = bias[16 + m];

  for (int tile = blockIdx.x * wpb + wave; tile < ntiles; tile += nwaves) {
    int rowA = tile * 16 + m;
    if (rowA >= N) rowA = 0;
    const float* xrow = x   + (size_t)rowA * 32;
    const float* grow = agg + (size_t)rowA * 32;

    v2f a[16];
#pragma unroll
    for (int kk = 0; kk < 16; ++kk) {
      const int k = kk * 4 + koff;   // k, k+1 never straddle the 32 boundary
      if (k < 32) { a[kk].x = xrow[k];      a[kk].y = xrow[k + 1]; }
      else        { a[kk].x = grow[k - 32]; a[kk].y = grow[k - 31]; }
    }
    v8f acc0 = {0.f,0.f,0.f,0.f,0.f,0.f,0.f,0.f};
    v8f acc1 = {0.f,0.f,0.f,0.f,0.f,0.f,0.f,0.f};
#pragma unroll
    for (int kk = 0; kk < 16; ++kk) {
      const v2f b0 = *(const v2f*)(ldsW + 2 * (kk * 64 + lane));
      const v2f b1 = *(const v2f*)(ldsW + 2 * (kk * 64 + 32 + lane));
      acc0 = __builtin_amdgcn_wmma_f32_16x16x4_f32(false, a[kk], false, b0, (short)0, acc0, false, false);
      acc1 = __builtin_amdgcn_wmma_f32_16x16x4_f32(false, a[kk], false, b1, (short)0, acc1, false, false);
    }
#pragma unroll
    for (int v = 0; v < 8; ++v) {
      const int node = tile * 16 + v + mrow;
      if (node < N) {
        xout[(size_t)node * 32 + m]      = lrelu(acc0[v] + bn0);
        xout[(size_t)node * 32 + 16 + m] = lrelu(acc1[v] + bn1);
      }
    }
  }
}

__global__ void gpn_zero_kernel(float4* __restrict__ p, int n4) {
  int i = blockIdx.x * blockDim.x + threadIdx.x;
  const int stride = gridDim.x * blockDim.x;
  const float4 z = make_float4(0.f, 0.f, 0.f, 0.f);
  for (; i < n4; i += stride) p[i] = z;
}

// Column sums of x(N,32) into sums[32]
__global__ void gpn_colsum_kernel(const float* __restrict__ x,
                                  float* __restrict__ sums, int N) {
  const int tid = blockIdx.x * blockDim.x + threadIdx.x;
  const int col = tid & 31;
  int row = tid >> 5;
  const int rstride = (gridDim.x * blockDim.x) >> 5;
  float s = 0.f;
  for (; row < N; row += rstride) s += x[(size_t)row * 32 + col];
  unsafeAtomicAdd(&sums[col], s);
}

// Single block: value head + policy logits on actionable nodes + softmax
__global__ void __launch_bounds__(1024)
gpn_head_kernel(const float* __restrict__ x,
                const float* __restrict__ sums,
                const int* __restrict__ act,
                const float* __restrict__ Wval, const float* __restrict__ bval,
                const float* __restrict__ Wpol, const float* __restrict__ bpol,
                float* __restrict__ out, int N, int A) {
  __shared__ float red[1024];
  const int t = threadIdx.x;
  float logit = -INFINITY;
  if (t < A) {
    int node = act[t];
    if ((unsigned)node >= (unsigned)N) node = 0;
    const float* row = x + (size_t)node * 32;
    float s = 0.f;
#pragma unroll
    for (int c = 0; c < 32; ++c) s += row[c] * Wpol[c];
    logit = s + bpol[0];
  }
  red[t] = logit;
  __syncthreads();
  for (int o = 512; o > 0; o >>= 1) {
    if (t < o) red[t] = fmaxf(red[t], red[t + o]);
    __syncthreads();
  }
  const float mx = red[0];
  __syncthreads();
  const float ex = (t < A) ? __expf(logit - mx) : 0.f;
  red[t] = ex;
  __syncthreads();
  for (int o = 512; o > 0; o >>= 1) {
    if (t < o) red[t] += red[t + o];
    __syncthreads();
  }
  const float denom = red[0];
  if (t < A) out[1 + t] = ex / denom;
  if (t == 0) {
    const float inv = 1.0f / (float)N;
    float v = 0.f;
#pragma unroll
    for (int c = 0; c < 32; ++c) v += (sums[c] * inv) * Wval[c];
    out[0] = v + bval[0];
  }
}

static inline int imin_host(int a, int b) { return a < b ? a : b; }

extern "C" void kernel_launch(void* const* d_in, const int* in_sizes, int n_in,
                              void* d_out, int out_size, void* d_ws, size_t ws_size,
                              hipStream_t stream) {
  const float* node_feats = (const float*)d_in[0];
  const int*   edge_index = (const int*)d_in[1];
  const int*   actionable = (const int*)d_in[2];
  const float* W_embed = (const float*)d_in[3];
  const float* b_embed = (const float*)d_in[4];
  const float* W_msg   = (const float*)d_in[5];
  const float* b_msg   = (const float*)d_in[6];
  const float* W_upd   = (const float*)d_in[7];
  const float* b_upd   = (const float*)d_in[8];
  const float* W_val   = (const float*)d_in[9];
  const float* b_val   = (const float*)d_in[10];
  const float* W_pol   = (const float*)d_in[11];
  const float* b_pol   = (const float*)d_in[12];

  const int N = in_sizes[0] / 128;
  const int E = in_sizes[1] / 2;
  const int A = in_sizes[2];

  float* xA   = (float*)d_ws;
  float* xB   = xA + (size_t)N * 32;
  float* agg  = xB + (size_t)N * 32;
  float* sums = agg + (size_t)N * 32;

  // 1) node embedding
  {
    const int ntiles = (N + 15) / 16;
    const int blocks = imin_host((ntiles + 7) / 8, 4096);
    gpn_embed_kernel<<<blocks, 256, 0, stream>>>(node_feats, W_embed, b_embed, xA, N);
  }

  // 2) five message-passing iterations (ping-pong xA/xB)
  float* x = xA;
  float* xn = xB;
  for (int it = 0; it < 5; ++it) {
    gpn_zero_kernel<<<1024, 256, 0, stream>>>((float4*)agg, N * 32 / 4);
    {
      const int etiles = (E + 15) / 16;
      const int blocks = imin_host((etiles + 7) / 8, 8192);
      gpn_msg_kernel<<<blocks, 256, 0, stream>>>(x, edge_index, edge_index + E,
                                                 W_msg, b_msg, agg, E, N);
    }
    {
      const int ntiles = (N + 15) / 16;
      const int blocks = imin_host((ntiles + 7) / 8, 4096);
      gpn_upd_kernel<<<blocks, 256, 0, stream>>>(x, agg, W_upd, b_upd, xn, N);
    }
    float* tmp = x; x = xn; xn = tmp;
  }

  // 3) heads
  gpn_zero_kernel<<<1, 32, 0, stream>>>((float4*)sums, 8);
  gpn_colsum_kernel<<<512, 256, 0, stream>>>(x, sums, N);
  gpn_head_kernel<<<1, 1024, 0, stream>>>(x, sums, actionable, W_val, b_val,
                                          W_pol, b_pol, (float*)d_out, N, A);
}